// IoULoss_44435731644976
// MI455X (gfx1250) — compile-verified
//
#include <hip/hip_runtime.h>

// IoU coverage loss on MI455X (gfx1250).
// Pipeline (all on `stream`, scratch in d_ws):
//   1. zero 4016x4016 f32 grid G (64.5 MB; fits in 192MB L2)
//   2. scatter +-1 at gt corners (JAX semantics: wrap negative idx, drop OOB)
//   3. row scan (axis 1, contiguous)           -> cumsum_y(D)
//   4. column scan via WMMA, fused (>0)        -> occ
//   5. row scan                                 -> cumsum_y(occ)
//   6. column scan via WMMA                     -> P = inclusive 2D SAT of occ
//   7. gather 4 corners per pred, block-reduce (1-iou)
//   8. deterministic final reduce -> d_out[0]
// Column scan = L16x16(lower-tri ones) x tile, done as 4 chained
// v_wmma_f32_16x16x4_f32 (exact: integers < 2^24).

#define GRIDN 4008
#define WPAD  4016   // 251 * 16, padded so 16x16 tiles divide evenly
#define NTILE 251

typedef float v2f __attribute__((ext_vector_type(2)));
typedef float v8f __attribute__((ext_vector_type(8)));

__global__ void zero_grid(float4* __restrict__ g, int n4) {
  int i = blockIdx.x * blockDim.x + threadIdx.x;
  if (i < n4) g[i] = make_float4(0.f, 0.f, 0.f, 0.f);
}

__global__ void scatter_boxes(const float* __restrict__ gt, float* __restrict__ G, int n) {
  int i = blockIdx.x * blockDim.x + threadIdx.x;
  if (i >= n) return;
  int x1 = __float2int_rn(gt[i * 4 + 0] * 100.0f);
  int y1 = __float2int_rn(gt[i * 4 + 1] * 100.0f);
  int x2 = __float2int_rn(gt[i * 4 + 2] * 100.0f);
  int y2 = __float2int_rn(gt[i * 4 + 3] * 100.0f);
  auto upd = [&](int x, int y, float v) {
    if (x < 0) x += GRIDN;                      // JAX: negative indices wrap
    if (y < 0) y += GRIDN;
    if (x >= 0 && x < GRIDN && y >= 0 && y < GRIDN)   // JAX: OOB scatter dropped
      atomicAdd(&G[(size_t)x * WPAD + y], v);   // integer-valued f32 add: exact, order-free
  };
  upd(x1, y1, 1.f);
  upd(x1, y2 + 1, -1.f);
  upd(x2 + 1, y1, -1.f);
  upd(x2 + 1, y2 + 1, 1.f);
}

// Inclusive prefix sum along the contiguous (y) dimension, one block per row.
__global__ void scan_rows(float* __restrict__ G) {
  __shared__ float s[256];
  const int t = threadIdx.x;
  float* R = G + (size_t)blockIdx.x * WPAD;
  const int base = t * 16;
  float v[16];
  float sum = 0.f;
#pragma unroll
  for (int j = 0; j < 16; ++j) {
    float x = (base + j < WPAD) ? R[base + j] : 0.f;
    sum += x;
    v[j] = sum;
  }
  s[t] = sum;
  __syncthreads();
  for (int st = 1; st < 256; st <<= 1) {
    float add = (t >= st) ? s[t - st] : 0.f;
    __syncthreads();
    s[t] += add;
    __syncthreads();
  }
  float off = s[t] - sum;  // exclusive offset for this thread's chunk
#pragma unroll
  for (int j = 0; j < 16; ++j)
    if (base + j < WPAD) R[base + j] = v[j] + off;
}

// Inclusive prefix sum down the columns (axis 0) of a 16-wide stripe.
// One wave (32 threads) per stripe; serial over 251 row-tiles with carry.
// Per tile: T = L * X via 4 chained v_wmma_f32_16x16x4_f32.
// If THRESH: store (T+carry > 0) ? 1 : 0 (carry update uses raw sums).
template <bool THRESH>
__global__ void scan_cols_wmma(float* __restrict__ G) {
  const int l  = threadIdx.x;   // 0..31, full wave, EXEC all ones
  const int n  = l & 15;        // column within stripe / row index for A
  const int hi = l >> 4;
  const size_t col0 = (size_t)blockIdx.x * 16;

  // A chunks: columns 4c..4c+3 of lower-triangular ones L (L[m][k] = k<=m).
  // 32-bit A 16x4 layout: lane m=l&15; VGPR j holds K = 2*hi + j.
  v2f A[4];
#pragma unroll
  for (int c = 0; c < 4; ++c)
#pragma unroll
    for (int j = 0; j < 2; ++j) {
      int k = 4 * c + 2 * hi + j;
      A[c][j] = (k <= n) ? 1.f : 0.f;
    }

  float carry = 0.f;
  float* base = G + col0 + n;   // per-lane column pointer

  for (int t = 0; t < NTILE; ++t) {
    float* tb = base + (size_t)t * 16 * WPAD;
    if (t + 1 < NTILE)
      __builtin_prefetch(tb + (size_t)16 * WPAD, 0, 0);  // global_prefetch_b8

    v8f acc = {0.f, 0.f, 0.f, 0.f, 0.f, 0.f, 0.f, 0.f};
#pragma unroll
    for (int c = 0; c < 4; ++c) {
      // B 4x16 layout: lane n=l&15; VGPR j holds K = 2*hi + j (row 4c+2*hi+j).
      v2f B;
      B[0] = tb[(size_t)(4 * c + 2 * hi + 0) * WPAD];
      B[1] = tb[(size_t)(4 * c + 2 * hi + 1) * WPAD];
      acc = __builtin_amdgcn_wmma_f32_16x16x4_f32(
          /*neg_a=*/false, A[c], /*neg_b=*/false, B,
          /*c_mod=*/(short)0, acc, /*reuse_a=*/false, /*reuse_b=*/false);
    }
    // Add running per-column carry (every lane's elements share N = l&15).
#pragma unroll
    for (int v = 0; v < 8; ++v) acc[v] += carry;
    // New carry = row M=15 of (T + carry): VGPR 7 on lanes 16..31.
    carry = __shfl(acc[7], 16 + n, 32);
    // Store: C/D layout: VGPR v -> row M = v + 8*hi, col N = l&15.
#pragma unroll
    for (int v = 0; v < 8; ++v) {
      float val = acc[v];
      if (THRESH) val = (val > 0.f) ? 1.f : 0.f;
      tb[(size_t)(v + 8 * hi) * WPAD] = val;
    }
  }
}

__global__ void gather_reduce(const float* __restrict__ pred, const float* __restrict__ P,
                              float* __restrict__ partials, int n) {
  __shared__ float s[256];
  int i = blockIdx.x * blockDim.x + threadIdx.x;
  float val = 0.f;
  if (i < n) {
    int x1 = __float2int_rn(pred[i * 4 + 0] * 100.0f);
    int y1 = __float2int_rn(pred[i * 4 + 1] * 100.0f);
    int x2 = __float2int_rn(pred[i * 4 + 2] * 100.0f);
    int y2 = __float2int_rn(pred[i * 4 + 3] * 100.0f);
    auto I = [&](int a, int b) -> float {
      a = min(max(a, 0), GRIDN);
      b = min(max(b, 0), GRIDN);
      return (a > 0 && b > 0) ? P[(size_t)(a - 1) * WPAD + (b - 1)] : 0.f;
    };
    float cov = I(x2, y2) - I(x1, y2) - I(x2, y1) + I(x1, y1);
    float area = (float)((x2 - x1) * (y2 - y1));
    bool valid = (x2 > x1) && (y2 > y1);
    float iou = valid ? cov / fmaxf(area, 1.f) : 0.f;
    val = 1.f - iou;
  }
  s[threadIdx.x] = val;
  __syncthreads();
  for (int st = 128; st > 0; st >>= 1) {
    if (threadIdx.x < st) s[threadIdx.x] += s[threadIdx.x + st];
    __syncthreads();
  }
  if (threadIdx.x == 0) partials[blockIdx.x] = s[0];
}

__global__ void final_reduce(const float* __restrict__ partials, int nb,
                             float* __restrict__ out, int n) {
  if (threadIdx.x == 0 && blockIdx.x == 0) {
    float s = 0.f;
    for (int i = 0; i < nb; ++i) s += partials[i];  // fixed order: deterministic
    out[0] = s / (float)n;
  }
}

extern "C" void kernel_launch(void* const* d_in, const int* in_sizes, int n_in,
                              void* d_out, int out_size, void* d_ws, size_t ws_size,
                              hipStream_t stream) {
  const float* pred = (const float*)d_in[0];
  // d_in[1] = target: unused by the reference loss
  const float* gt = (const float*)d_in[2];
  float* out = (float*)d_out;

  float* G = (float*)d_ws;                       // 4016*4016 f32 = 64.5 MB
  const size_t gridElems = (size_t)WPAD * WPAD;
  float* partials = G + gridElems;               // small partial-sum buffer

  const int npred = in_sizes[0] / 4;
  const int ngt   = in_sizes[2] / 4;
  const int n4 = (int)(gridElems / 4);

  zero_grid<<<(n4 + 255) / 256, 256, 0, stream>>>((float4*)G, n4);
  scatter_boxes<<<(ngt + 255) / 256, 256, 0, stream>>>(gt, G, ngt);
  scan_rows<<<WPAD, 256, 0, stream>>>(G);                 // cumsum_y(D)
  scan_cols_wmma<true><<<NTILE, 32, 0, stream>>>(G);      // cumsum_x, fused >0 -> occ
  scan_rows<<<WPAD, 256, 0, stream>>>(G);                 // cumsum_y(occ)
  scan_cols_wmma<false><<<NTILE, 32, 0, stream>>>(G);     // cumsum_x -> SAT
  const int nb = (npred + 255) / 256;
  gather_reduce<<<nb, 256, 0, stream>>>(pred, G, partials, npred);
  final_reduce<<<1, 32, 0, stream>>>(partials, nb, out, npred);
}